// ComplexGraphAttention_23545010716679
// MI455X (gfx1250) — compile-verified
//
#include <hip/hip_runtime.h>
#include <hip/hip_bf16.h>

typedef float v2f __attribute__((ext_vector_type(2)));
typedef float v8f __attribute__((ext_vector_type(8)));

#define HIDDEN 128
#define HEADS 4
#define HEAD_DIM 32
#define NEG_FLT_MAX (-3.402823466e38f)

// ---------- float <-> order-preserving unsigned encoding (for atomic max) ----
__device__ __forceinline__ unsigned encodeF(float f) {
    unsigned u = __float_as_uint(f);
    return (u & 0x80000000u) ? ~u : (u | 0x80000000u);
}
__device__ __forceinline__ float decodeF(unsigned e) {
    return (e & 0x80000000u) ? __uint_as_float(e ^ 0x80000000u)
                             : __uint_as_float(~e);
}

// ---------- K0: init workspace ----------------------------------------------
__global__ void init_ws_kernel(float* __restrict__ att, size_t n_att,
                               unsigned* __restrict__ hmax, float* __restrict__ hsum) {
    size_t i = (size_t)blockIdx.x * blockDim.x + threadIdx.x;
    size_t stride = (size_t)gridDim.x * blockDim.x;
    for (; i < n_att; i += stride) att[i] = 0.0f;
    if (blockIdx.x == 0 && threadIdx.x < HEADS) {
        hmax[threadIdx.x] = 0u;       // below encoding of every real value
        hsum[threadIdx.x] = 0.0f;
    }
}

// ---------- K1: fused QKV projection via V_WMMA_F32_16X16X4_F32 -------------
// One wave computes a 16-node tile of Q,K,V (16 x 128 each).
// A (x tile, 16x4): lanes 0-15 hold K=0,1 ; lanes 16-31 hold K=2,3 (2 VGPRs).
// B (W.T tile, 4x16): mirrored layout, column = n0 + (lane&15).
// D (16x16): lane -> N column, vgpr j -> M row (halves split M 0-7 / 8-15).
__global__ void qkv_wmma_kernel(const float* __restrict__ x,
                                const float* __restrict__ Wq, const float* __restrict__ bq,
                                const float* __restrict__ Wk, const float* __restrict__ bk,
                                const float* __restrict__ Wv, const float* __restrict__ bv,
                                float* __restrict__ Q, float* __restrict__ K,
                                float* __restrict__ V, int n_nodes) {
    const int wave = (int)((blockIdx.x * blockDim.x + threadIdx.x) >> 5);
    const int lane = threadIdx.x & 31;
    const int row0 = wave * 16;
    if (row0 >= n_nodes) return;          // whole-wave exit: EXEC stays all-1s

    const int half = lane >> 4;           // 0: lanes 0-15, 1: lanes 16-31
    const int l    = lane & 15;

    // A-matrix fragment: x[row0+l][half*2 + {0,1}]
    const float* xr = x + (size_t)(row0 + l) * 4 + half * 2;
    v2f a; a.x = xr[0]; a.y = xr[1];

    #pragma unroll
    for (int t = 0; t < 8; ++t) {
        const int ncol = t * 16 + l;      // output channel this lane owns
        #pragma unroll
        for (int m = 0; m < 3; ++m) {
            const float* W  = (m == 0) ? Wq : (m == 1) ? Wk : Wv;
            const float* bb = (m == 0) ? bq : (m == 1) ? bk : bv;
            float*       O  = (m == 0) ? Q  : (m == 1) ? K  : V;

            // B-matrix fragment: W[ncol][half*2 + {0,1}]  (W is [128][4] row-major)
            v2f b;
            b.x = W[ncol * 4 + half * 2 + 0];
            b.y = W[ncol * 4 + half * 2 + 1];

            float bias = bb[ncol];
            v8f c;
            #pragma unroll
            for (int j = 0; j < 8; ++j) c[j] = bias;

            c = __builtin_amdgcn_wmma_f32_16x16x4_f32(
                    false, a, false, b, (short)0, c, false, false);

            #pragma unroll
            for (int j = 0; j < 8; ++j)
                O[(size_t)(row0 + half * 8 + j) * HIDDEN + ncol] = c[j];
        }
    }
}

// ---------- K2: per-edge scores + global per-head max ------------------------
__global__ void score_kernel(const int* __restrict__ row, const int* __restrict__ col,
                             const float* __restrict__ Q, const float* __restrict__ K,
                             float* __restrict__ scores, unsigned* __restrict__ hmax,
                             int n_edges) {
    const int e = blockIdx.x * blockDim.x + threadIdx.x;
    float s[HEADS] = {NEG_FLT_MAX, NEG_FLT_MAX, NEG_FLT_MAX, NEG_FLT_MAX};
    if (e < n_edges) {
        const int r = row[e], c = col[e];
        const float4* qr = (const float4*)(Q + (size_t)r * HIDDEN);
        const float4* kc = (const float4*)(K + (size_t)c * HIDDEN);
        #pragma unroll
        for (int h = 0; h < HEADS; ++h) {
            float acc = 0.0f;
            #pragma unroll
            for (int i = 0; i < HEAD_DIM / 4; ++i) {
                float4 q = qr[h * (HEAD_DIM / 4) + i];
                float4 k = kc[h * (HEAD_DIM / 4) + i];
                acc += q.x * k.x + q.y * k.y + q.z * k.z + q.w * k.w;
            }
            s[h] = acc * 0.17677669529663687f;   // 1/sqrt(32)
            scores[(size_t)e * HEADS + h] = s[h];
        }
    }
    __shared__ float red[256];
    #pragma unroll
    for (int h = 0; h < HEADS; ++h) {
        red[threadIdx.x] = s[h];
        __syncthreads();
        for (int off = 128; off > 0; off >>= 1) {
            if ((int)threadIdx.x < off)
                red[threadIdx.x] = fmaxf(red[threadIdx.x], red[threadIdx.x + off]);
            __syncthreads();
        }
        if (threadIdx.x == 0) atomicMax(&hmax[h], encodeF(red[0]));
        __syncthreads();
    }
}

// ---------- K3: exp(s - max) in place + global per-head sum ------------------
__global__ void expsum_kernel(float* __restrict__ scores,
                              const unsigned* __restrict__ hmax,
                              float* __restrict__ hsum, int n_edges) {
    __shared__ float mx[HEADS];
    if (threadIdx.x < HEADS) mx[threadIdx.x] = decodeF(hmax[threadIdx.x]);
    __syncthreads();

    const int e = blockIdx.x * blockDim.x + threadIdx.x;
    float v[HEADS] = {0.0f, 0.0f, 0.0f, 0.0f};
    if (e < n_edges) {
        #pragma unroll
        for (int h = 0; h < HEADS; ++h) {
            v[h] = __expf(scores[(size_t)e * HEADS + h] - mx[h]);
            scores[(size_t)e * HEADS + h] = v[h];
        }
    }
    __shared__ float red[256];
    #pragma unroll
    for (int h = 0; h < HEADS; ++h) {
        red[threadIdx.x] = v[h];
        __syncthreads();
        for (int off = 128; off > 0; off >>= 1) {
            if ((int)threadIdx.x < off)
                red[threadIdx.x] += red[threadIdx.x + off];
            __syncthreads();
        }
        if (threadIdx.x == 0) atomicAdd(&hsum[h], red[0]);
        __syncthreads();
    }
}

// ---------- K4: scatter attended[col] += (w/sum) * V[row] --------------------
// One work-item handles 4 hidden dims of one edge (32 work-items / edge).
__global__ void scatter_kernel(const int* __restrict__ row, const int* __restrict__ col,
                               const float* __restrict__ V, const float* __restrict__ w,
                               const float* __restrict__ hsum,
                               float* __restrict__ att, int n_edges) {
    const size_t t = (size_t)blockIdx.x * blockDim.x + threadIdx.x;
    const size_t e = t >> 5;
    if (e >= (size_t)n_edges) return;
    const int sub = (int)(t & 31);
    const int i0  = sub * 4;              // hidden dims [i0, i0+3]
    const int h   = i0 >> 5;              // head index (32 dims/head)

    const int r = row[e], c = col[e];
    const float wn = w[e * HEADS + h] / hsum[h];
    const float4 v = *(const float4*)(V + (size_t)r * HIDDEN + i0);
    float* dst = att + (size_t)c * HIDDEN + i0;
    atomicAdd(dst + 0, wn * v.x);
    atomicAdd(dst + 1, wn * v.y);
    atomicAdd(dst + 2, wn * v.z);
    atomicAdd(dst + 3, wn * v.w);
}

// ---------- K5: out = attended @ Wo.T + bo + x -------------------------------
__global__ void out_kernel(const float* __restrict__ att, const float* __restrict__ x,
                           const float* __restrict__ Wo, const float* __restrict__ bo,
                           float* __restrict__ out, int n_nodes) {
    const int i = blockIdx.x * blockDim.x + threadIdx.x;
    if (i >= n_nodes) return;
    const float4* a = (const float4*)(att + (size_t)i * HIDDEN);
    float acc[4] = {bo[0], bo[1], bo[2], bo[3]};
    for (int j = 0; j < HIDDEN / 4; ++j) {
        const float4 av = a[j];
        #pragma unroll
        for (int o = 0; o < 4; ++o) {
            const float4 wv = ((const float4*)(Wo + o * HIDDEN))[j];
            acc[o] += av.x * wv.x + av.y * wv.y + av.z * wv.z + av.w * wv.w;
        }
    }
    #pragma unroll
    for (int o = 0; o < 4; ++o)
        out[(size_t)i * 4 + o] = acc[o] + x[(size_t)i * 4 + o];
}

// ---------- launch -----------------------------------------------------------
extern "C" void kernel_launch(void* const* d_in, const int* in_sizes, int n_in,
                              void* d_out, int out_size, void* d_ws, size_t ws_size,
                              hipStream_t stream) {
    const float* x   = (const float*)d_in[0];
    const int*   ei  = (const int*)d_in[1];
    const float* Wq  = (const float*)d_in[2];
    const float* bq  = (const float*)d_in[3];
    const float* Wk  = (const float*)d_in[4];
    const float* bk  = (const float*)d_in[5];
    const float* Wv  = (const float*)d_in[6];
    const float* bv  = (const float*)d_in[7];
    const float* Wo  = (const float*)d_in[8];
    const float* bo  = (const float*)d_in[9];
    float* out = (float*)d_out;

    const int n_nodes = in_sizes[0] / 4;
    const int n_edges = in_sizes[1] / 2;
    const int* row = ei;
    const int* col = ei + n_edges;

    // workspace layout (floats)
    float* ws = (float*)d_ws;
    const size_t nQ = (size_t)n_nodes * HIDDEN;
    float*    Q     = ws;
    float*    K     = ws + nQ;
    float*    V     = ws + 2 * nQ;
    float*    att   = ws + 3 * nQ;
    float*    sc    = ws + 4 * nQ;                         // [E, HEADS]
    unsigned* hmax  = (unsigned*)(ws + 4 * nQ + (size_t)n_edges * HEADS);
    float*    hsum  = (float*)(hmax + HEADS);

    // K0: init attended + per-head max/sum
    init_ws_kernel<<<2048, 256, 0, stream>>>(att, nQ, hmax, hsum);

    // K1: QKV projection (one wave per 16-node tile, 8 waves/block)
    {
        const int tiles  = (n_nodes + 15) / 16;
        const int blocks = (tiles + 7) / 8;
        qkv_wmma_kernel<<<blocks, 256, 0, stream>>>(x, Wq, bq, Wk, bk, Wv, bv,
                                                    Q, K, V, n_nodes);
    }

    // K2: edge scores + global max
    score_kernel<<<(n_edges + 255) / 256, 256, 0, stream>>>(row, col, Q, K, sc,
                                                            hmax, n_edges);

    // K3: exp + global sum
    expsum_kernel<<<(n_edges + 255) / 256, 256, 0, stream>>>(sc, hmax, hsum, n_edges);

    // K4: scatter (32 work-items per edge)
    {
        const size_t total  = (size_t)n_edges * 32;
        const size_t blocks = (total + 255) / 256;
        scatter_kernel<<<(unsigned)blocks, 256, 0, stream>>>(row, col, V, sc, hsum,
                                                             att, n_edges);
    }

    // K5: output projection + residual
    out_kernel<<<(n_nodes + 255) / 256, 256, 0, stream>>>(att, x, Wo, bo, out, n_nodes);
}